// AA_encoder_40278203302163
// MI455X (gfx1250) — compile-verified
//
#include <hip/hip_runtime.h>
#include <hip/hip_bf16.h>

// ---------------------------------------------------------------------------
// Problem constants (match reference)
// ---------------------------------------------------------------------------
#define B_SZ   512
#define L_MAX  48
#define T_LEN  128
#define DIM    300
#define N_A    2048
#define N_C    2048
#define HEADS  4
#define DK     75
#define TOPK   96
#define ML     (B_SZ * L_MAX)          // 24576 rows of the big GEMMs

typedef float v2f __attribute__((ext_vector_type(2)));
typedef float v8f __attribute__((ext_vector_type(8)));

// ---------------------------------------------------------------------------
// 0) zero workspace region
// ---------------------------------------------------------------------------
__global__ void zero_kernel(float* __restrict__ p, size_t n) {
    size_t i = (size_t)blockIdx.x * blockDim.x + threadIdx.x;
    size_t stride = (size_t)gridDim.x * blockDim.x;
    for (; i < n; i += stride) p[i] = 0.0f;
}

// ---------------------------------------------------------------------------
// 1) aa_embed[n,d] = sum_t input[map_AA[n],t,d]*mask[n,t]; scatter into inner
// ---------------------------------------------------------------------------
__global__ __launch_bounds__(320) void aspect_embed_kernel(
    const float* __restrict__ input_embed, const float* __restrict__ mask,
    const int* __restrict__ mapAA, const int* __restrict__ mapAAidx,
    float* __restrict__ inner)
{
    int n = blockIdx.x;
    __shared__ float s_mask[T_LEN];
    for (int t = threadIdx.x; t < T_LEN; t += 320) s_mask[t] = mask[n * T_LEN + t];
    __syncthreads();
    int d = threadIdx.x;
    if (d < DIM) {
        int m = mapAA[n];
        const float* base = input_embed + (size_t)m * T_LEN * DIM + d;
        float acc = 0.0f;
        #pragma unroll 8
        for (int t = 0; t < T_LEN; ++t) acc += base[(size_t)t * DIM] * s_mask[t];
        inner[((size_t)m * L_MAX + mapAAidx[n]) * DIM + d] = acc;
    }
}

// ---------------------------------------------------------------------------
// 2) scatter clause rows into inner
// ---------------------------------------------------------------------------
__global__ __launch_bounds__(320) void clause_scatter_kernel(
    const float* __restrict__ clause, const int* __restrict__ mapAS,
    const int* __restrict__ mapASidx, float* __restrict__ inner)
{
    int n = blockIdx.x;
    int d = threadIdx.x;
    if (d < DIM)
        inner[((size_t)mapAS[n] * L_MAX + mapASidx[n]) * DIM + d] = clause[(size_t)n * DIM + d];
}

// ---------------------------------------------------------------------------
// 3) C[m,n] = sum_k A[m,k]*Bw[n,k] + bias[n]   (NT GEMM, fp32 WMMA 16x16x4)
//    optional epilogue: C = max(C,0)/denom[m]
//    grid = (M/16, ceil(N/16)), block = 32 (one wave per 16x16 tile)
// ---------------------------------------------------------------------------
__global__ __launch_bounds__(32) void gemm_nt_wmma(
    const float* __restrict__ A, const float* __restrict__ Bw,
    const float* __restrict__ bias, const float* __restrict__ denom,
    float* __restrict__ C, int N, int K, int lda, int ldb, int ldc, int do_relu)
{
    int m0 = blockIdx.x * 16;
    int n0 = blockIdx.y * 16;
    int lane = threadIdx.x;
    int h = lane >> 4;          // 0/1 : K-half selector
    int l = lane & 15;          // row (A) / col (B) within tile
    int nn = n0 + l;
    int ncl = nn < N ? nn : (N - 1);          // clamp: garbage cols never stored
    const float* ap = A  + (size_t)(m0 + l) * lda + 2 * h;
    const float* bp = Bw + (size_t)ncl      * ldb + 2 * h;
    v8f acc = {};
    for (int k = 0; k < K; k += 4) {
        v2f a = *(const v2f*)(ap + k);        // (K=2h, 2h+1) for this lane
        v2f b = *(const v2f*)(bp + k);
        acc = __builtin_amdgcn_wmma_f32_16x16x4_f32(
                  false, a, false, b, (short)0, acc, false, false);
    }
    if (nn < N) {
        float bv = bias[nn];
        #pragma unroll
        for (int r = 0; r < 8; ++r) {
            int row = m0 + r + 8 * h;
            float v = acc[r] + bv;
            if (do_relu) v = fmaxf(v, 0.0f) / denom[row];
            C[(size_t)row * ldc + nn] = v;
        }
    }
}

// ---------------------------------------------------------------------------
// 4) batched NN GEMM: C[b,m,n] = sum_k A[b,m,k]*B[b,k,n]  (adj @ inner)
//    grid = (M/16, ceil(N/16), batch), block = 32
// ---------------------------------------------------------------------------
__global__ __launch_bounds__(32) void gemm_nn_wmma_batched(
    const float* __restrict__ Abase, const float* __restrict__ Bbase,
    float* __restrict__ Cbase, int N, int K, int lda, int ldb, int ldc,
    size_t strideA, size_t strideB, size_t strideC)
{
    int b = blockIdx.z;
    const float* A = Abase + (size_t)b * strideA;
    const float* B = Bbase + (size_t)b * strideB;
    float*       C = Cbase + (size_t)b * strideC;
    int m0 = blockIdx.x * 16;
    int n0 = blockIdx.y * 16;
    int lane = threadIdx.x;
    int h = lane >> 4;
    int l = lane & 15;
    int nn = n0 + l;
    int ncl = nn < N ? nn : (N - 1);
    const float* ap = A + (size_t)(m0 + l) * lda + 2 * h;
    v8f acc = {};
    for (int k = 0; k < K; k += 4) {
        v2f a = *(const v2f*)(ap + k);
        int kk = k + 2 * h;
        v2f bv;
        bv.x = B[(size_t)kk       * ldb + ncl];
        bv.y = B[(size_t)(kk + 1) * ldb + ncl];
        acc = __builtin_amdgcn_wmma_f32_16x16x4_f32(
                  false, a, false, bv, (short)0, acc, false, false);
    }
    if (nn < N) {
        #pragma unroll
        for (int r = 0; r < 8; ++r)
            C[(size_t)(m0 + r + 8 * h) * ldc + nn] = acc[r];
    }
}

// ---------------------------------------------------------------------------
// 5) per-batch: scores -> softmax -> head mean -> diag/row mask ->
//    exact top-96 threshold (binary search on float bits, values >= 0) ->
//    sel + sel^T reweight -> rowsum denominators
//    1 block per batch element, 256 threads, ~55 KB LDS
// ---------------------------------------------------------------------------
__global__ __launch_bounds__(256) void attn_adj_kernel(
    const float* __restrict__ q, const float* __restrict__ k,
    const int* __restrict__ lengths,
    float* __restrict__ adj_out, float* __restrict__ denom_out)
{
    int b = blockIdx.x;
    int tid = threadIdx.x;
    __shared__ float s_attn[HEADS * L_MAX * L_MAX];   // 36864 B
    __shared__ float s_adj[L_MAX * L_MAX];            //  9216 B
    __shared__ float s_fin[L_MAX * L_MAX];            //  9216 B
    __shared__ int   s_cnt;

    int len = lengths[b];

    // ---- phase 1: scores + softmax per (h,i) row ----
    if (tid < HEADS * L_MAX) {
        int h = tid / L_MAX, i = tid % L_MAX;
        const float* qrow = q + ((size_t)b * L_MAX + i) * DIM + h * DK;
        float qreg[DK];
        #pragma unroll
        for (int d = 0; d < DK; ++d) qreg[d] = qrow[d];
        float* srow = &s_attn[(h * L_MAX + i) * L_MAX];
        const float inv_sqrt_dk = 0.11547005383792516f;   // 1/sqrt(75)
        float mx = -3.0e38f;
        for (int j = 0; j < L_MAX; ++j) {
            const float* krow = k + ((size_t)b * L_MAX + j) * DIM + h * DK;
            float s = 0.0f;
            #pragma unroll 5
            for (int d = 0; d < DK; ++d) s += qreg[d] * krow[d];
            s *= inv_sqrt_dk;
            if (j >= len) s = -1.0e9f;
            srow[j] = s;
            mx = fmaxf(mx, s);
        }
        float se = 0.0f;
        for (int j = 0; j < L_MAX; ++j) { float e = __expf(srow[j] - mx); srow[j] = e; se += e; }
        float inv = 1.0f / se;
        for (int j = 0; j < L_MAX; ++j) srow[j] *= inv;
    }
    __syncthreads();

    // ---- phase 2: head mean, diag=1, zero padded rows ----
    for (int e = tid; e < L_MAX * L_MAX; e += 256) {
        int i = e / L_MAX, j = e % L_MAX;
        float a = 0.25f * (s_attn[(0 * L_MAX + i) * L_MAX + j] +
                           s_attn[(1 * L_MAX + i) * L_MAX + j] +
                           s_attn[(2 * L_MAX + i) * L_MAX + j] +
                           s_attn[(3 * L_MAX + i) * L_MAX + j]);
        if (i == j)  a = 1.0f;
        if (i >= len) a = 0.0f;
        s_adj[e] = a;
    }
    __syncthreads();

    // ---- phase 3: exact kth-largest via binary search on uint bits ----
    // all values are non-negative floats -> bit order == value order
    unsigned lo = 0u, hi = 0x40000000u;   // 2.0f strictly exceeds every entry
    while (hi - lo > 1u) {
        unsigned mid = lo + ((hi - lo) >> 1);
        if (tid == 0) s_cnt = 0;
        __syncthreads();
        int c = 0;
        for (int e = tid; e < L_MAX * L_MAX; e += 256)
            if (__float_as_uint(s_adj[e]) >= mid) ++c;
        atomicAdd(&s_cnt, c);
        __syncthreads();
        int cnt = s_cnt;
        if (cnt >= TOPK) lo = mid; else hi = mid;
        __syncthreads();
    }
    float kth = __uint_as_float(lo);

    // ---- phase 4: sel = (a>=kth); sel+sel^T (0/1/2!); diag=1; reweight ----
    for (int e = tid; e < L_MAX * L_MAX; e += 256) {
        int i = e / L_MAX, j = e % L_MAX;
        float aij = s_adj[e];
        float selv;
        if (i == j) selv = 1.0f;
        else selv = (float)((aij >= kth) + (s_adj[j * L_MAX + i] >= kth));
        float fin = selv * aij;
        s_fin[e] = fin;
        adj_out[(size_t)b * L_MAX * L_MAX + e] = fin;
    }
    __syncthreads();

    // ---- phase 5: denom[i] = rowsum + 1 ----
    if (tid < L_MAX) {
        float s = 0.0f;
        for (int j = 0; j < L_MAX; ++j) s += s_fin[tid * L_MAX + j];
        denom_out[(size_t)b * L_MAX + tid] = s + 1.0f;
    }
}

// ---------------------------------------------------------------------------
// 6) final gather: clause rows from gcn, fallback when graph has <=1 node
// ---------------------------------------------------------------------------
__global__ __launch_bounds__(320) void final_gather_kernel(
    const float* __restrict__ gcn, const float* __restrict__ clause,
    const int* __restrict__ mapAS, const int* __restrict__ mapASidx,
    const int* __restrict__ lengths, float* __restrict__ out)
{
    int n = blockIdx.x;
    int d = threadIdx.x;
    if (d >= DIM) return;
    int b = mapAS[n];
    int i = mapASidx[n];
    float v = (lengths[b] > 1)
            ? gcn[((size_t)b * L_MAX + i) * DIM + d]
            : clause[(size_t)n * DIM + d];
    out[(size_t)n * DIM + d] = v;
}

// ---------------------------------------------------------------------------
// launch
// ---------------------------------------------------------------------------
extern "C" void kernel_launch(void* const* d_in, const int* in_sizes, int n_in,
                              void* d_out, int out_size, void* d_ws, size_t ws_size,
                              hipStream_t stream) {
    const float* input_embed = (const float*)d_in[0];
    const float* clause      = (const float*)d_in[1];
    const float* aa_mask     = (const float*)d_in[2];
    const int*   lengths     = (const int*)  d_in[3];
    const int*   mapAA       = (const int*)  d_in[4];
    const int*   mapAAidx    = (const int*)  d_in[5];
    const int*   mapAS       = (const int*)  d_in[6];
    const int*   mapASidx    = (const int*)  d_in[7];
    const float* Wq          = (const float*)d_in[8];
    const float* bq          = (const float*)d_in[9];
    const float* Wk          = (const float*)d_in[10];
    const float* bk          = (const float*)d_in[11];
    const float* Wg          = (const float*)d_in[12];
    const float* bg          = (const float*)d_in[13];
    float* out = (float*)d_out;

    // workspace layout (floats): ~93.3 MB total
    const size_t SZ_INNER = (size_t)ML * DIM;          // 7,372,800
    float* ws    = (float*)d_ws;
    float* inner = ws;                                 // [B*L, D]
    float* qbuf  = inner + SZ_INNER;                   // q, later ax = adj@inner
    float* kbuf  = qbuf  + SZ_INNER;                   // k, later gcn
    float* adjb  = kbuf  + SZ_INNER;                   // [B, L, L]
    float* denb  = adjb  + (size_t)B_SZ * L_MAX * L_MAX; // [B, L]

    const int MT = ML / 16;                 // 1536 M-tiles
    const int NT = (DIM + 15) / 16;         // 19 N-tiles

    zero_kernel<<<1024, 256, 0, stream>>>(inner, SZ_INNER);
    aspect_embed_kernel<<<N_A, 320, 0, stream>>>(input_embed, aa_mask, mapAA, mapAAidx, inner);
    clause_scatter_kernel<<<N_C, 320, 0, stream>>>(clause, mapAS, mapASidx, inner);

    // Q = inner @ Wq^T + bq ; K = inner @ Wk^T + bk     (fp32 WMMA)
    gemm_nt_wmma<<<dim3(MT, NT), 32, 0, stream>>>(inner, Wq, bq, nullptr, qbuf,
                                                  DIM, DIM, DIM, DIM, DIM, 0);
    gemm_nt_wmma<<<dim3(MT, NT), 32, 0, stream>>>(inner, Wk, bk, nullptr, kbuf,
                                                  DIM, DIM, DIM, DIM, DIM, 0);

    // scores/softmax/top-k adjacency + denominators
    attn_adj_kernel<<<B_SZ, 256, 0, stream>>>(qbuf, kbuf, lengths, adjb, denb);

    // ax = adj @ inner   (batched fp32 WMMA; overwrites qbuf, which is dead)
    gemm_nn_wmma_batched<<<dim3(L_MAX / 16, NT, B_SZ), 32, 0, stream>>>(
        adjb, inner, qbuf, DIM, L_MAX, L_MAX, DIM, DIM,
        (size_t)L_MAX * L_MAX, (size_t)L_MAX * DIM, (size_t)L_MAX * DIM);

    // gcn = relu(ax @ Wg^T + bg) / denom   (into kbuf, which is dead)
    gemm_nt_wmma<<<dim3(MT, NT), 32, 0, stream>>>(qbuf, Wg, bg, denb, kbuf,
                                                  DIM, DIM, DIM, DIM, DIM, 1);

    final_gather_kernel<<<N_C, 320, 0, stream>>>(kbuf, clause, mapAS, mapASidx, lengths, out);
}